// DifferentiableNLMS_9466107921234
// MI455X (gfx1250) — compile-verified
//
#include <hip/hip_runtime.h>

// Problem constants (from reference)
#define B_N 32
#define T_N 2000
#define F_N 513
#define L_N 32          // FILTER_LENGTH
#define STEPSZ 0.1f
#define EPSV 1e-8f

// Tiling
#define TS 32                            // time steps per LDS tile (== L: static circular indexing)
#define FC 64                            // f-channels per block (2 wave32s)
#define FULL_TILES 62                    // tiles 0..61 are fully inside T=2000
#define TAIL_STEPS (T_N - FULL_TILES*TS) // 16

typedef __attribute__((address_space(1))) int as1_int;  // global
typedef __attribute__((address_space(3))) int as3_int;  // LDS
typedef unsigned int v4u __attribute__((ext_vector_type(4)));
typedef int v8i __attribute__((ext_vector_type(8)));
typedef int v4i __attribute__((ext_vector_type(4)));

// ---------- CDNA5 waits ----------
__device__ __forceinline__ void wait_tensor0() {
#if __has_builtin(__builtin_amdgcn_s_wait_tensorcnt)
  __builtin_amdgcn_s_wait_tensorcnt(0);
#else
  asm volatile("s_wait_tensorcnt 0" ::: "memory");
#endif
}
__device__ __forceinline__ void wait_async0() {
#if __has_builtin(__builtin_amdgcn_s_wait_asynccnt)
  __builtin_amdgcn_s_wait_asynccnt(0);
#else
  asm volatile("s_wait_asynccnt 0" ::: "memory");
#endif
}

// ---------- LDS byte address of a __shared__ object ----------
__device__ __forceinline__ unsigned lds_addr_of(void* p) {
  return (unsigned)(size_t)(as3_int*)p;
}

// ---------- Tile staging: TDM (preferred) or per-lane async (fallback) ----------
#if __has_builtin(__builtin_amdgcn_tensor_load_to_lds)
#define USE_TDM 1
// 2D TDM load: tile_dim0 = FC floats per row, tile_dim1 = TS rows,
// row stride F_N. OOB (col >= tensor_dim0 or row >= tensor_dim1) reads zero,
// which cleanly handles both the f-tail block and the t-tail rows.
__device__ __forceinline__ void tdm_tile_load(const float* gtile, unsigned lds_addr,
                                              int tdim0, int tdim1) {
  unsigned long long ga = (unsigned long long)(size_t)gtile;
  v4u g0;
  g0[0] = 1u;                                   // count=1, user mode, no gather
  g0[1] = lds_addr;                             // lds_addr [63:32]
  g0[2] = (unsigned)ga;                         // global_addr lo
  g0[3] = (unsigned)(ga >> 32) | (2u << 30);    // global_addr hi + type=2 ("image")
  v8i g1;
  g1[0] = (int)(2u << 16);                                              // data_size=4B
  g1[1] = (int)(((unsigned)tdim0 & 0xFFFFu) << 16);                     // tensor_dim0 lo
  g1[2] = (int)(((unsigned)tdim0 >> 16) | (((unsigned)tdim1 & 0xFFFFu) << 16)); // d0 hi | d1 lo
  g1[3] = (int)(((unsigned)tdim1 >> 16) | ((unsigned)FC << 16));        // d1 hi | tile_dim0=64
  g1[4] = TS;                                                           // tile_dim1=32, tile_dim2=0
  g1[5] = F_N;                                                          // tensor_dim0_stride=513
  g1[6] = 0;
  g1[7] = 0;
  v4i z4 = {0, 0, 0, 0};
  v8i z8 = {0, 0, 0, 0, 0, 0, 0, 0};
  __builtin_amdgcn_tensor_load_to_lds(g0, g1, z4, z4, z8, 0);
}
#else
#define USE_TDM 0
__device__ __forceinline__ void async_cp_b32(const float* g, float* l) {
#if __has_builtin(__builtin_amdgcn_global_load_async_to_lds_b32)
  __builtin_amdgcn_global_load_async_to_lds_b32((as1_int*)g, (as3_int*)l, 0, 0);
#else
  *l = *g;
#endif
}
#endif

// ---------- One NLMS step (jj must be a compile-time constant in unrolled loops) ----
// 8 independent FMA chains of depth 4 for the y_hat reduction: minimizes the
// loop-carried critical path at ~1 wave/SIMD occupancy and feeds VOPD dual-issue.
#define NLMS_STEP(jj)                                                        \
  {                                                                          \
    const float x = sX[buf][(jj)][tid];                                      \
    const float yv = sY[buf][(jj)][tid];                                     \
    const float xold = win[(jj) & (L_N - 1)];                                \
    win[(jj) & (L_N - 1)] = x;                                               \
    S += x * x - xold * xold;                                                \
    float a0 = 0.f, a1 = 0.f, a2 = 0.f, a3 = 0.f;                            \
    float a4 = 0.f, a5 = 0.f, a6 = 0.f, a7 = 0.f;                            \
    _Pragma("unroll") for (int l = 0; l < 4; ++l) {                          \
      a0 += W[l + 0]  * win[((jj) + l + 1)  & (L_N - 1)];                    \
      a1 += W[l + 4]  * win[((jj) + l + 5)  & (L_N - 1)];                    \
      a2 += W[l + 8]  * win[((jj) + l + 9)  & (L_N - 1)];                    \
      a3 += W[l + 12] * win[((jj) + l + 13) & (L_N - 1)];                    \
      a4 += W[l + 16] * win[((jj) + l + 17) & (L_N - 1)];                    \
      a5 += W[l + 20] * win[((jj) + l + 21) & (L_N - 1)];                    \
      a6 += W[l + 24] * win[((jj) + l + 25) & (L_N - 1)];                    \
      a7 += W[l + 28] * win[((jj) + l + 29) & (L_N - 1)];                    \
    }                                                                        \
    const float yh = ((a0 + a1) + (a2 + a3)) + ((a4 + a5) + (a6 + a7));      \
    const float e = yv - yh;                                                 \
    const float g = STEPSZ * e * __builtin_amdgcn_rcpf(S + EPSV);            \
    _Pragma("unroll") for (int l = 0; l < L_N; ++l)                          \
      W[l] += g * win[((jj) + l + 1) & (L_N - 1)];                           \
    *eptr = e;                                                               \
    eptr += estride;                                                         \
  }

__global__ __launch_bounds__(FC) void nlms_scan_kernel(
    const float* __restrict__ X,      // [B, T, F]
    const float* __restrict__ Y,      // [B, T, F]
    const float* __restrict__ Wprev,  // [B, L, F]
    float* __restrict__ Eout,         // [B, T, F]
    float* __restrict__ Wout,         // [B, L, F]
    float* __restrict__ wsDummy) {    // sink for out-of-range lanes' stores
  __shared__ float sX[2][TS][FC];   // 16 KB
  __shared__ float sY[2][TS][FC];   // 16 KB

  const int tid = threadIdx.x;
  const int b = blockIdx.y;
  const int f0 = blockIdx.x * FC;
  const int f = f0 + tid;
  const bool valid = (f < F_N);

  // Filter taps + circular window live in VGPRs.
  float W[L_N], win[L_N];
  float S = 0.0f;
#pragma unroll
  for (int l = 0; l < L_N; ++l) { win[l] = 0.0f; W[l] = 0.0f; }
  if (valid) {
#pragma unroll
    for (int l = 0; l < L_N; ++l)
      W[l] = Wprev[((size_t)b * L_N + l) * F_N + f];
  }

  const size_t baseBF = (size_t)b * T_N * F_N + (size_t)f;
  const float* Xtile0 = X + (size_t)b * T_N * F_N + f0;  // block-uniform tile origin
  const float* Ytile0 = Y + (size_t)b * T_N * F_N + f0;

  // Branchless E-store: invalid lanes write to the scratch sink with stride 0.
  float* eptr = valid ? (Eout + baseBF) : wsDummy;
  const size_t estride = valid ? (size_t)F_N : 0;

  // ---- Prologue: stage tile 0 into buffer 0 (wave0 -> X, wave1 -> Y) ----
#if USE_TDM
  if (tid < 32) {
    tdm_tile_load(Xtile0, lds_addr_of(&sX[0][0][0]), F_N - f0, T_N);
  } else {
    tdm_tile_load(Ytile0, lds_addr_of(&sY[0][0][0]), F_N - f0, T_N);
  }
  wait_tensor0();
#else
  if (valid) {
#pragma unroll
    for (int j = 0; j < TS; ++j) {
      async_cp_b32(X + baseBF + (size_t)j * F_N, &sX[0][j][tid]);
      async_cp_b32(Y + baseBF + (size_t)j * F_N, &sY[0][j][tid]);
    }
  }
  wait_async0();
#endif
  __syncthreads();

  int buf = 0;
#pragma unroll 1
  for (int tile = 0; tile < FULL_TILES; ++tile) {
    const int kb2 = (tile + 1) * TS;  // next tile start, always <= 1984
    const int nb = buf ^ 1;

    // ---- Issue DMA for the next tile; overlaps with compute below ----
#if USE_TDM
    if (tid < 32) {
      tdm_tile_load(Xtile0 + (size_t)kb2 * F_N, lds_addr_of(&sX[nb][0][0]),
                    F_N - f0, T_N - kb2);   // OOB rows/cols -> zeros
    } else {
      tdm_tile_load(Ytile0 + (size_t)kb2 * F_N, lds_addr_of(&sY[nb][0][0]),
                    F_N - f0, T_N - kb2);
    }
#else
    if (valid) {
#pragma unroll
      for (int j = 0; j < TS; ++j) {
        const int kj = (kb2 + j < T_N) ? (kb2 + j) : (T_N - 1);  // branchless clamp
        async_cp_b32(X + baseBF + (size_t)kj * F_N, &sX[nb][j][tid]);
        async_cp_b32(Y + baseBF + (size_t)kj * F_N, &sY[nb][j][tid]);
      }
    }
#endif

    // ---- 32 NLMS steps, no guards (all tiles here are full) ----
#pragma unroll
    for (int j = 0; j < TS; ++j) { NLMS_STEP(j) }

#if USE_TDM
    wait_tensor0();
#else
    wait_async0();
#endif
    __syncthreads();
    buf ^= 1;
  }

  // ---- Epilogue: final partial tile (k = 1984..1999), statically 16 steps ----
#pragma unroll
  for (int j = 0; j < TAIL_STEPS; ++j) { NLMS_STEP(j) }

  // ---- Final filter state ----
  if (valid) {
#pragma unroll
    for (int l = 0; l < L_N; ++l)
      Wout[((size_t)b * L_N + l) * F_N + f] = W[l];
  }
}

extern "C" void kernel_launch(void* const* d_in, const int* in_sizes, int n_in,
                              void* d_out, int out_size, void* d_ws, size_t ws_size,
                              hipStream_t stream) {
  const float* X = (const float*)d_in[0];      // X_hat_mag [B,T,F]
  const float* Y = (const float*)d_in[1];      // Y_mag     [B,T,F]
  const float* Wp = (const float*)d_in[2];     // W_prev    [B,L,F]
  float* E = (float*)d_out;                                    // E_hat_mag [B,T,F]
  float* Wf = (float*)d_out + (size_t)B_N * T_N * F_N;         // W_final   [B,L,F]

  dim3 grid((F_N + FC - 1) / FC, B_N, 1);  // (9, 32)
  nlms_scan_kernel<<<grid, FC, 0, stream>>>(X, Y, Wp, E, Wf, (float*)d_ws);
}